// GraphEncoder_53867479826897
// MI455X (gfx1250) — compile-verified
//
#include <hip/hip_runtime.h>
#include <math.h>

static constexpr float NEG_SLOPE = 0.2f;
static constexpr float EPS_F     = 1e-16f;

typedef __attribute__((ext_vector_type(2))) float v2f;
typedef __attribute__((ext_vector_type(8))) float v8f;

// ---------- helpers ----------
__device__ __forceinline__ unsigned fenc(float f) {
  unsigned b = __float_as_uint(f);
  return (b & 0x80000000u) ? ~b : (b | 0x80000000u);   // monotonic float->uint
}
__device__ __forceinline__ float fdec(unsigned k) {
  return (k & 0x80000000u) ? __uint_as_float(k & 0x7FFFFFFFu) : __uint_as_float(~k);
}
__device__ __forceinline__ float lrelu(float v) { return v > 0.f ? v : v * NEG_SLOPE; }

__device__ __forceinline__ void edge_sd(const long long* __restrict__ ei, long long E,
                                        long long e, int& s, int& d) {
  if (e < E) { s = (int)ei[e]; d = (int)ei[E + e]; }
  else       { s = d = (int)(e - E); }                 // appended self-loops
}

// async copy of one 16B chunk: global memory at base plus goff to the LDS address in ldsOff, ASYNCcnt-tracked
__device__ __forceinline__ void async_g2l_b128(unsigned ldsOff, unsigned goff,
                                               unsigned long long base) {
  asm volatile("global_load_async_to_lds_b128 %0, %1, %2"
               :: "v"(ldsOff), "v"(goff), "s"(base) : "memory");
}
__device__ __forceinline__ void wait_async0() {
  asm volatile("s_wait_asynccnt 0x0" ::: "memory");
}

// ---------- fp32 WMMA GEMM, LDS-staged: C[M,N] = A[M,K] @ B[K,N] ----------
// One block (8 waves) per 16-row strip of C. Async-stages the full B panel
// (K*N floats) + the block's A strip (16*K floats) into LDS, then each wave
// computes 16x16 tiles with V_WMMA_F32_16X16X4_F32 reading fragments from LDS.
__global__ void wmma_gemm_f32_lds(const float* __restrict__ A,
                                  const float* __restrict__ B,
                                  float* __restrict__ C, int M, int N, int K) {
  extern __shared__ float smem[];          // [K*N] B panel, then [16*K] A strip
  float* Bs = smem;
  float* As = smem + (size_t)K * N;

  const int tid   = threadIdx.x;           // 256 threads = 8 waves
  const int row0  = blockIdx.x << 4;

  // ---- stage B panel (K*N*4 bytes) ----
  {
    unsigned ldsBase = (unsigned)(uintptr_t)(void*)Bs;
    int nchunk = (K * N) >> 2;             // 16B chunks
    unsigned long long gbase = (unsigned long long)(uintptr_t)B;
    for (int c = tid; c < nchunk; c += blockDim.x)
      async_g2l_b128(ldsBase + c * 16u, (unsigned)(c * 16), gbase);
  }
  // ---- stage A strip (16*K*4 bytes) ----
  {
    unsigned ldsBase = (unsigned)(uintptr_t)(void*)As;
    int nchunk = (16 * K) >> 2;
    unsigned long long gbase =
        (unsigned long long)(uintptr_t)(A + (size_t)row0 * K);
    for (int c = tid; c < nchunk; c += blockDim.x)
      async_g2l_b128(ldsBase + c * 16u, (unsigned)(c * 16), gbase);
  }
  wait_async0();
  __syncthreads();

  const int wave   = tid >> 5;
  const int nwaves = blockDim.x >> 5;
  const int lane   = tid & 31;
  const int half   = lane >> 4;            // 0 -> K pair {0,1}, 1 -> {2,3}
  const int lid    = lane & 15;
  const int ntiles = N >> 4;

  for (int nt = wave; nt < ntiles; nt += nwaves) {
    const int col0 = nt << 4;
    v8f acc = {};
    for (int kb = 0; kb < K; kb += 4) {
      // A frag: lanes 0-15 -> M=lid K=kb+0/1 ; lanes 16-31 -> M=lid K=kb+2/3
      v2f a;
      a.x = As[lid * K + kb + half * 2 + 0];
      a.y = As[lid * K + kb + half * 2 + 1];
      // B frag: VGPR0 -> K=kb+2*half, VGPR1 -> K=kb+2*half+1, N=col0+lid
      v2f b;
      b.x = Bs[(size_t)(kb + half * 2 + 0) * N + col0 + lid];
      b.y = Bs[(size_t)(kb + half * 2 + 1) * N + col0 + lid];
      acc = __builtin_amdgcn_wmma_f32_16x16x4_f32(false, a, false, b,
                                                  (short)0, acc, false, false);
    }
    // C layout: VGPR r -> (M=row0+r+8*half, N=col0+lid)
#pragma unroll
    for (int r = 0; r < 8; ++r)
      C[(size_t)(row0 + r + half * 8) * N + col0 + lid] = acc[r];
  }
}

// ---------- per-(node,head) attention coefficients ----------
__global__ void att_coeffs(const float* __restrict__ xw,
                           const float* __restrict__ att_src,
                           const float* __restrict__ att_dst,
                           float* __restrict__ a_src, float* __restrict__ a_dst,
                           int N, int H, int C) {
  int i = blockIdx.x * blockDim.x + threadIdx.x;
  if (i >= N * H) return;
  int n = i / H, h = i % H;
  const float* row = xw + (size_t)n * H * C + (size_t)h * C;
  const float* as = att_src + h * C;
  const float* ad = att_dst + h * C;
  float s = 0.f, d = 0.f;
  for (int c = 0; c < C; ++c) { s += row[c] * as[c]; d += row[c] * ad[c]; }
  a_src[i] = s; a_dst[i] = d;
}

// ---------- edge pass 1: segment max of leaky-relu logits ----------
__global__ void edge_max(const long long* __restrict__ ei, long long E, int N,
                         const float* __restrict__ a_src, const float* __restrict__ a_dst,
                         unsigned* __restrict__ m_enc, int H) {
  long long Et = E + N;
  long long i = blockIdx.x * (long long)blockDim.x + threadIdx.x;
  if (i >= Et * H) return;
  long long e = i / H; int h = (int)(i % H);
  int s, d; edge_sd(ei, E, e, s, d);
  float v = lrelu(a_src[(size_t)s * H + h] + a_dst[(size_t)d * H + h]);
  atomicMax(&m_enc[(size_t)d * H + h], fenc(v));
}

// ---------- edge pass 2: exp(e - max), accumulate denominator ----------
__global__ void edge_exp(const long long* __restrict__ ei, long long E, int N,
                         const float* __restrict__ a_src, const float* __restrict__ a_dst,
                         const unsigned* __restrict__ m_enc,
                         float* __restrict__ denom, float* __restrict__ alpha, int H) {
  long long Et = E + N;
  long long i = blockIdx.x * (long long)blockDim.x + threadIdx.x;
  if (i >= Et * H) return;
  long long e = i / H; int h = (int)(i % H);
  int s, d; edge_sd(ei, E, e, s, d);
  float v  = lrelu(a_src[(size_t)s * H + h] + a_dst[(size_t)d * H + h]);
  float ex = __expf(v - fdec(m_enc[(size_t)d * H + h]));
  alpha[i] = ex;
  atomicAdd(&denom[(size_t)d * H + h], ex);
}

// ---------- edge pass 3: normalize alpha ----------
__global__ void edge_norm(const long long* __restrict__ ei, long long E, int N,
                          const float* __restrict__ denom, float* __restrict__ alpha,
                          int H) {
  long long Et = E + N;
  long long i = blockIdx.x * (long long)blockDim.x + threadIdx.x;
  if (i >= Et * H) return;
  long long e = i / H; int h = (int)(i % H);
  int s, d; edge_sd(ei, E, e, s, d);
  alpha[i] = alpha[i] / (denom[(size_t)d * H + h] + EPS_F);
}

// ---------- edge aggregation (float4): agg[dst, c..c+3] += alpha * xw[src, c..c+3] ----------
__global__ void edge_aggregate4(const long long* __restrict__ ei, long long E, int N,
                                const float* __restrict__ alpha,
                                const float* __restrict__ xw,
                                float* __restrict__ agg, int H, int C) {
  const int Ct  = H * C;
  const int Cq  = Ct >> 2;                         // float4 groups per row
  long long Et  = E + N;
  long long i   = blockIdx.x * (long long)blockDim.x + threadIdx.x;
  if (i >= Et * (long long)Cq) return;
  long long e = i / Cq; int q = (int)(i % Cq);
  int c = q << 2;
  int h = c / C;
  int s, d; edge_sd(ei, E, e, s, d);
  float a = alpha[e * H + h];
  float4 v = *(const float4*)(xw + (size_t)s * Ct + c);
  float* dst = agg + (size_t)d * Ct + c;
  atomicAdd(dst + 0, a * v.x);
  atomicAdd(dst + 1, a * v.y);
  atomicAdd(dst + 2, a * v.z);
  atomicAdd(dst + 3, a * v.w);
}

// ---------- bias + ELU (in place) ----------
__global__ void bias_elu(float* __restrict__ agg, const float* __restrict__ bias,
                         int N, int Ct) {
  long long i = blockIdx.x * (long long)blockDim.x + threadIdx.x;
  if (i >= (long long)N * Ct) return;
  int c = (int)(i % Ct);
  float v = agg[i] + bias[c];
  agg[i] = v > 0.f ? v : (__expf(v) - 1.f);
}

// ---------- final: out[c] = b2[c] (init) ----------
__global__ void init_out(float* __restrict__ out, const float* __restrict__ b2, int C) {
  int c = threadIdx.x;
  if (c < C) out[c] = b2[c];
}

// ---------- final: out[c] += mean_n agg2[n, c] ----------
__global__ void mean_reduce(const float* __restrict__ agg2, float* __restrict__ out,
                            int N, int C) {
  int c = threadIdx.x;                    // blockDim.x == C
  float partial = 0.f;
  for (int n = blockIdx.x; n < N; n += gridDim.x) {
    __builtin_prefetch(agg2 + (size_t)(n + 4 * gridDim.x) * C + c, 0, 0);
    partial += agg2[(size_t)n * C + c];
  }
  atomicAdd(&out[c], partial / (float)N);
}

// =====================================================================
extern "C" void kernel_launch(void* const* d_in, const int* in_sizes, int n_in,
                              void* d_out, int out_size, void* d_ws, size_t ws_size,
                              hipStream_t stream) {
  const float*     x        = (const float*)d_in[0];
  const long long* ei       = (const long long*)d_in[1];   // int64 [2, E]
  const float*     W1       = (const float*)d_in[2];
  const float*     att_src1 = (const float*)d_in[3];
  const float*     att_dst1 = (const float*)d_in[4];
  const float*     b1       = (const float*)d_in[5];
  const float*     W2       = (const float*)d_in[6];
  const float*     att_src2 = (const float*)d_in[7];
  const float*     att_dst2 = (const float*)d_in[8];
  const float*     b2       = (const float*)d_in[9];
  float*           out      = (float*)d_out;

  const int  IN = 128, HID = 32, HEADS = 4, OUT = 64;
  const int  C1 = HEADS * HID;          // 128
  const int  N  = in_sizes[0] / IN;     // 50000
  const long long E  = in_sizes[1] / 2; // 1,600,000
  const long long Et = E + N;

  // ---- workspace carve (256B aligned) ----
  char* w = (char*)d_ws;
  auto carve = [&](size_t bytes) {
    char* p = w; w += (bytes + 255) & ~(size_t)255; return p;
  };
  float*    xw    = (float*)carve((size_t)N * C1 * 4);  // xw1, then xw2 (N*64)
  float*    agg1  = (float*)carve((size_t)N * C1 * 4);  // agg1 -> h (in place)
  float*    agg2  = (float*)carve((size_t)N * OUT * 4);
  float*    asrc  = (float*)carve((size_t)N * HEADS * 4);
  float*    adst  = (float*)carve((size_t)N * HEADS * 4);
  unsigned* menc  = (unsigned*)carve((size_t)N * HEADS * 4);
  float*    denom = (float*)carve((size_t)N * HEADS * 4);
  float*    alpha = (float*)carve((size_t)Et * HEADS * 4);
  (void)ws_size; (void)n_in; (void)out_size;

  const int B = 256;
  auto cdiv = [](long long a, long long b) { return (unsigned)((a + b - 1) / b); };

  // ================= Layer 1 (H=4, C=32, concat) =================
  {
    unsigned smem = (unsigned)((IN * C1 + 16 * IN) * sizeof(float)); // 72 KB
    wmma_gemm_f32_lds<<<N / 16, B, smem, stream>>>(x, W1, xw, N, C1, IN);

    att_coeffs<<<cdiv((long long)N * HEADS, B), B, 0, stream>>>(
        xw, att_src1, att_dst1, asrc, adst, N, HEADS, HID);

    hipMemsetAsync(menc,  0, (size_t)N * HEADS * 4, stream);   // == fenc(-inf) floor
    hipMemsetAsync(denom, 0, (size_t)N * HEADS * 4, stream);
    hipMemsetAsync(agg1,  0, (size_t)N * C1 * 4, stream);

    edge_max <<<cdiv(Et * HEADS, B), B, 0, stream>>>(ei, E, N, asrc, adst, menc, HEADS);
    edge_exp <<<cdiv(Et * HEADS, B), B, 0, stream>>>(ei, E, N, asrc, adst, menc, denom, alpha, HEADS);
    edge_norm<<<cdiv(Et * HEADS, B), B, 0, stream>>>(ei, E, N, denom, alpha, HEADS);
    edge_aggregate4<<<cdiv(Et * (long long)(C1 / 4), B), B, 0, stream>>>(
        ei, E, N, alpha, xw, agg1, HEADS, HID);

    bias_elu<<<cdiv((long long)N * C1, B), B, 0, stream>>>(agg1, b1, N, C1);
  }

  // ================= Layer 2 (H=1, C=64, mean==identity) =================
  {
    unsigned smem = (unsigned)((C1 * OUT + 16 * C1) * sizeof(float)); // 40 KB
    wmma_gemm_f32_lds<<<N / 16, B, smem, stream>>>(agg1, W2, xw, N, OUT, C1);

    att_coeffs<<<cdiv((long long)N, B), B, 0, stream>>>(
        xw, att_src2, att_dst2, asrc, adst, N, 1, OUT);

    hipMemsetAsync(menc,  0, (size_t)N * 4, stream);
    hipMemsetAsync(denom, 0, (size_t)N * 4, stream);
    hipMemsetAsync(agg2,  0, (size_t)N * OUT * 4, stream);

    edge_max <<<cdiv(Et, B), B, 0, stream>>>(ei, E, N, asrc, adst, menc, 1);
    edge_exp <<<cdiv(Et, B), B, 0, stream>>>(ei, E, N, asrc, adst, menc, denom, alpha, 1);
    edge_norm<<<cdiv(Et, B), B, 0, stream>>>(ei, E, N, denom, alpha, 1);
    edge_aggregate4<<<cdiv(Et * (long long)(OUT / 4), B), B, 0, stream>>>(
        ei, E, N, alpha, xw, agg2, 1, OUT);
  }

  // ================= final mean over nodes =================
  init_out<<<1, OUT, 0, stream>>>(out, b2, OUT);
  mean_reduce<<<256, OUT, 0, stream>>>(agg2, out, N, OUT);
}